// GCN3_MN_67980742361102
// MI455X (gfx1250) — compile-verified
//
#include <hip/hip_runtime.h>
#include <math.h>

// Problem constants (match reference file)
#define NN 50000      // nodes
#define NE 800000     // edges
#define NG 64         // graphs
#define HD 64         // hidden dim
#define INF_ 4        // input feature dim

typedef __attribute__((ext_vector_type(2))) float v2f;
typedef __attribute__((ext_vector_type(8))) float v8f;

// ---------------------------------------------------------------- utilities
__global__ void zero_kernel(float* __restrict__ p, int n) {
    int i = blockIdx.x * blockDim.x + threadIdx.x;
    if (i < n) p[i] = 0.0f;
}

// ---------------------------------------------------------------- degrees
__global__ void deg_kernel(const int* __restrict__ src, const int* __restrict__ dst,
                           float* __restrict__ in_deg, float* __restrict__ out_deg) {
    int e = blockIdx.x * blockDim.x + threadIdx.x;
    if (e < NE) {
        atomicAdd(&out_deg[src[e]], 1.0f);
        atomicAdd(&in_deg[dst[e]], 1.0f);
    }
}

// ------------------------------------------------- node features + norms
__global__ void node_init_kernel(const float* __restrict__ in_deg,
                                 const float* __restrict__ out_deg,
                                 float* __restrict__ h4,
                                 float* __restrict__ norm_src,
                                 float* __restrict__ norm_dst) {
    int n = blockIdx.x * blockDim.x + threadIdx.x;
    if (n < NN) {
        float id = in_deg[n];
        float od = out_deg[n];
        h4[n * 4 + 0] = id;
        h4[n * 4 + 1] = (id - 3.0f > 0.0f) ? 1.0f : 0.0f;
        h4[n * 4 + 2] = 3.0f / id;           // inf when id==0, matches JAX
        h4[n * 4 + 3] = (id - 4.0f > 0.0f) ? 1.0f : 0.0f;
        norm_src[n] = rsqrtf(fmaxf(od, 1.0f));
        norm_dst[n] = rsqrtf(fmaxf(id, 1.0f));
    }
}

// ------------------------------------------- edge scatter: m = (h*ns)[src] -> agg[dst]
// One thread per (edge, feature). Within a 64-thread group all lanes share one
// edge -> coalesced feature reads and coalesced L2 f32 atomics.
template <int F>
__global__ void scatter_kernel(const float* __restrict__ h,
                               const float* __restrict__ norm_src,
                               const int* __restrict__ src,
                               const int* __restrict__ dst,
                               float* __restrict__ agg) {
    long long tid = (long long)blockIdx.x * blockDim.x + threadIdx.x;
    long long e = tid / F;
    int f = (int)(tid % F);
    if (e < NE) {
        int s = src[e];
        int d = dst[e];
        float v = h[(long long)s * F + f] * norm_src[s];
        atomicAdd(&agg[(long long)d * F + f], v);
    }
}

// ----------------------------------------- WMMA GEMM: out = relu((agg*nd) @ W + b)
// Block = 128 threads = 4 wave32s. Block computes rows [16*bx,16*bx+16) x all 64 cols.
// Wave w owns cols [16w, 16w+16). K in {4, 64}, stepped by 4 with f32 WMMA.
template <int K>
__global__ __launch_bounds__(128) void gemm_bias_relu_kernel(
        const float* __restrict__ agg, const float* __restrict__ norm_dst,
        const float* __restrict__ W,   const float* __restrict__ bias,
        float* __restrict__ out) {
    __shared__ float sW[K * HD];          // W is (K x 64) row-major
    __shared__ float sA[16 * (K + 1)];    // A tile, padded to dodge bank conflicts

    const int tid  = threadIdx.x;
    const int row0 = blockIdx.x * 16;     // N % 16 == 0, no row guards needed

    for (int i = tid; i < K * HD; i += 128) sW[i] = W[i];
    for (int i = tid; i < 16 * K; i += 128) {
        int r = i / K, k = i % K;
        sA[r * (K + 1) + k] = agg[(long long)(row0 + r) * K + k] * norm_dst[row0 + r];
    }
    __syncthreads();

    const int lane = tid & 31;
    const int wave = tid >> 5;
    const int col0 = wave * 16;
    const int hi   = lane >> 4;           // 0: lanes 0-15, 1: lanes 16-31
    const int m    = lane & 15;

    v8f acc = {0.f, 0.f, 0.f, 0.f, 0.f, 0.f, 0.f, 0.f};

    #pragma unroll
    for (int k0 = 0; k0 < K; k0 += 4) {
        const int ka = k0 + hi * 2;       // ISA 16x4 f32 A/B lane layout
        v2f a, b;
        a.x = sA[m * (K + 1) + ka];
        a.y = sA[m * (K + 1) + ka + 1];
        b.x = sW[ka * HD + col0 + m];
        b.y = sW[(ka + 1) * HD + col0 + m];
        acc = __builtin_amdgcn_wmma_f32_16x16x4_f32(
                /*neg_a=*/false, a, /*neg_b=*/false, b,
                /*c_mod=*/(short)0, acc, /*reuse_a=*/false, /*reuse_b=*/false);
    }

    #pragma unroll
    for (int r = 0; r < 8; ++r) {         // C layout: VGPR r -> row r (+8 for hi half)
        int row = row0 + r + hi * 8;
        int col = col0 + m;
        float v = acc[r] + bias[col];
        out[(long long)row * HD + col] = fmaxf(v, 0.0f);
    }
}

// ---------------------------------------------------------------- pooling
__global__ void pool_kernel(const float* __restrict__ h, const int* __restrict__ gid,
                            float* __restrict__ emb, float* __restrict__ counts) {
    int t = blockIdx.x * blockDim.x + threadIdx.x;
    if (t < NN * HD) {
        int n = t >> 6;
        int f = t & 63;
        int g = gid[n];
        atomicAdd(&emb[g * HD + f], h[t]);
        if (f == 0) atomicAdd(&counts[g], 1.0f);
    }
}

__global__ void out_kernel(const float* __restrict__ emb, const float* __restrict__ counts,
                           const float* __restrict__ Wout, const float* __restrict__ bout,
                           float* __restrict__ out) {
    int g = threadIdx.x;                  // 64 threads, one per graph
    if (g < NG) {
        float inv = 1.0f / counts[g];
        float s = bout[0];
        #pragma unroll 8
        for (int k = 0; k < HD; ++k) s += emb[g * HD + k] * inv * Wout[k];
        out[g] = 1.0f / (1.0f + __expf(-s));
    }
}

// ---------------------------------------------------------------- launcher
extern "C" void kernel_launch(void* const* d_in, const int* in_sizes, int n_in,
                              void* d_out, int out_size, void* d_ws, size_t ws_size,
                              hipStream_t stream) {
    (void)in_sizes; (void)n_in; (void)out_size; (void)ws_size;

    const int*   src  = (const int*)d_in[0];
    const int*   dst  = (const int*)d_in[1];
    const int*   gid  = (const int*)d_in[2];
    const float* W1   = (const float*)d_in[3];
    const float* b1   = (const float*)d_in[4];
    const float* W2   = (const float*)d_in[5];
    const float* b2   = (const float*)d_in[6];
    const float* W3   = (const float*)d_in[7];
    const float* b3   = (const float*)d_in[8];
    const float* W4   = (const float*)d_in[9];
    const float* b4   = (const float*)d_in[10];
    const float* Wout = (const float*)d_in[11];
    const float* bout = (const float*)d_in[12];
    float*       out  = (float*)d_out;

    // workspace carve-up (floats)
    float* ws       = (float*)d_ws;
    float* in_deg   = ws;                       // N
    float* out_deg  = in_deg  + NN;             // N
    float* norm_src = out_deg + NN;             // N
    float* norm_dst = norm_src + NN;            // N
    float* hbuf     = norm_dst + NN;            // N*64 (holds N*4 for layer 1 input)
    float* aggbuf   = hbuf    + (long long)NN * HD;  // N*64
    float* emb      = aggbuf  + (long long)NN * HD;  // G*64
    float* counts   = emb     + NG * HD;        // G

    const int TPB = 256;
    auto cdiv = [](long long a, long long b) { return (int)((a + b - 1) / b); };

    // degrees
    zero_kernel<<<cdiv(2 * NN, TPB), TPB, 0, stream>>>(in_deg, 2 * NN);
    deg_kernel<<<cdiv(NE, TPB), TPB, 0, stream>>>(src, dst, in_deg, out_deg);
    node_init_kernel<<<cdiv(NN, TPB), TPB, 0, stream>>>(in_deg, out_deg, hbuf, norm_src, norm_dst);

    // layer 1 (F_in = 4)
    zero_kernel<<<cdiv((long long)NN * INF_, TPB), TPB, 0, stream>>>(aggbuf, NN * INF_);
    scatter_kernel<INF_><<<cdiv((long long)NE * INF_, TPB), TPB, 0, stream>>>(hbuf, norm_src, src, dst, aggbuf);
    gemm_bias_relu_kernel<INF_><<<NN / 16, 128, 0, stream>>>(aggbuf, norm_dst, W1, b1, hbuf);

    // layers 2..4 (F_in = 64)
    const float* Ws[3] = {W2, W3, W4};
    const float* bs[3] = {b2, b3, b4};
    for (int l = 0; l < 3; ++l) {
        zero_kernel<<<cdiv((long long)NN * HD, TPB), TPB, 0, stream>>>(aggbuf, NN * HD);
        scatter_kernel<HD><<<cdiv((long long)NE * HD, TPB), TPB, 0, stream>>>(hbuf, norm_src, src, dst, aggbuf);
        gemm_bias_relu_kernel<HD><<<NN / 16, 128, 0, stream>>>(aggbuf, norm_dst, Ws[l], bs[l], hbuf);
    }

    // mean pool + readout
    zero_kernel<<<cdiv(NG * HD + NG, TPB), TPB, 0, stream>>>(emb, NG * HD + NG);
    pool_kernel<<<cdiv((long long)NN * HD, TPB), TPB, 0, stream>>>(hbuf, gid, emb, counts);
    out_kernel<<<1, 64, 0, stream>>>(emb, counts, Wout, bout, out);
}